// FocalLoss_75995151335937
// MI455X (gfx1250) — compile-verified
//
#include <hip/hip_runtime.h>
#include <hip/hip_bf16.h>

// ---------------------------------------------------------------------------
// RetinaNet focal loss for MI455X (gfx1250, wave32).
// Bandwidth-bound: 307 MB single-use stream of classification probs.
// Block reductions use V_WMMA_F32_16X16X4_F32 with an all-ones A matrix:
//   D[m][n] = sum_k B[k][n]  (every row identical), so
//   total = (sum of all D entries) / 16, independent of VGPR lane layout.
// ---------------------------------------------------------------------------

typedef float v2f __attribute__((ext_vector_type(2)));
typedef float v8f __attribute__((ext_vector_type(8)));

#define TPB 256
#define MAXM 64

// Sum 256 per-thread partials (blockDim.x == 256). Result valid on tid==0.
__device__ __forceinline__ float block_sum_wmma(float v, float* s_part) {
    const int tid = threadIdx.x;
    s_part[tid] = v;
    __syncthreads();
    float r = 0.0f;
    if (tid < 32) {  // wave 0 only: uniform branch -> EXEC all-ones in this wave
        v2f a; a[0] = 1.0f; a[1] = 1.0f;   // A = ones(16x4)
        v8f c = {};                         // C starts at 0, accumulates chunks
#pragma unroll
        for (int chunk = 0; chunk < TPB; chunk += 64) {
            v2f bb;
            bb[0] = s_part[chunk + 2 * tid];
            bb[1] = s_part[chunk + 2 * tid + 1];
            c = __builtin_amdgcn_wmma_f32_16x16x4_f32(
                /*neg_a=*/false, a, /*neg_b=*/false, bb,
                /*c_mod=*/(short)0, c, /*reuse_a=*/false, /*reuse_b=*/false);
        }
        // Every row of D equals the column sums; sum of all 256 D entries = 16*total.
        r = c[0] + c[1] + c[2] + c[3] + c[4] + c[5] + c[6] + c[7];
        r += __shfl_xor(r, 1, 32);
        r += __shfl_xor(r, 2, 32);
        r += __shfl_xor(r, 4, 32);
        r += __shfl_xor(r, 8, 32);
        r += __shfl_xor(r, 16, 32);
        r *= (1.0f / 16.0f);
    }
    __syncthreads();
    return r;
}

__global__ void zero_kernel(float* __restrict__ acc, int n) {
    int i = blockIdx.x * blockDim.x + threadIdx.x;
    if (i < n) acc[i] = 0.0f;
}

__device__ __forceinline__ float smooth_l1(float d) {
    return (d <= (1.0f / 9.0f)) ? 4.5f * d * d : d - (0.5f / 9.0f);
}

// One thread per anchor. Computes IoU argmax vs M GT boxes (cached in LDS),
// writes packed state, accumulates n_pos and smooth-L1 regression sum.
__global__ __launch_bounds__(TPB)
void assign_kernel(const float* __restrict__ anchors,   // [A,4]
                   const float* __restrict__ regs,      // [B,A,4]
                   const float* __restrict__ annots,    // [B,M,5]
                   int* __restrict__ state,             // [B,A]
                   float* __restrict__ reg_sum,         // [B]
                   float* __restrict__ npos,            // [B]
                   int A, int M) {
    __shared__ float s_ann[MAXM * 5];
    __shared__ float s_part[TPB];
    const int b   = blockIdx.y;
    const int tid = threadIdx.x;
    const int a   = blockIdx.x * TPB + tid;
    const int ann_n = M * 5;
    for (int i = tid; i < ann_n; i += TPB)
        s_ann[i] = annots[(size_t)b * ann_n + i];
    __syncthreads();

    float posf = 0.0f, rsum = 0.0f;
    if (a < A) {
        const float ax1 = anchors[a * 4 + 0], ay1 = anchors[a * 4 + 1];
        const float ax2 = anchors[a * 4 + 2], ay2 = anchors[a * 4 + 3];
        const float aw = ax2 - ax1, ah = ay2 - ay1;
        const float acx = ax1 + 0.5f * aw, acy = ay1 + 0.5f * ah;
        const float a_area = aw * ah;

        float best = -2.0f; int arg = 0;
        for (int m = 0; m < M; ++m) {
            const float* g = &s_ann[m * 5];
            float v = -1.0f;
            if (g[4] >= 0.0f) {
                float garea = (g[2] - g[0]) * (g[3] - g[1]);
                float iw = fmaxf(fminf(ax2, g[2]) - fmaxf(ax1, g[0]), 0.0f);
                float ih = fmaxf(fminf(ay2, g[3]) - fmaxf(ay1, g[1]), 0.0f);
                float inter = iw * ih;
                float uni = fmaxf(a_area + garea - inter, 1e-6f);
                v = inter / uni;
            }
            if (v > best) { best = v; arg = m; }  // first-max (argmax) semantics
        }
        const float* g = &s_ann[arg * 5];
        const bool pos = best >= 0.5f;
        const bool neg = best < 0.4f;
        int st = pos ? (int)g[4] : (neg ? -1 : -2);
        state[(size_t)b * A + a] = st;

        if (pos) {
            posf = 1.0f;
            float gw  = fmaxf(g[2] - g[0], 1.0f);
            float gh  = fmaxf(g[3] - g[1], 1.0f);
            float gcx = g[0] + 0.5f * (g[2] - g[0]);
            float gcy = g[1] + 0.5f * (g[3] - g[1]);
            float t0 = ((gcx - acx) / aw) / 0.1f;
            float t1 = ((gcy - acy) / ah) / 0.1f;
            float t2 = logf(gw / aw) / 0.2f;
            float t3 = logf(gh / ah) / 0.2f;
            const float* r = &regs[((size_t)b * A + a) * 4];
            rsum = smooth_l1(fabsf(t0 - r[0])) + smooth_l1(fabsf(t1 - r[1])) +
                   smooth_l1(fabsf(t2 - r[2])) + smooth_l1(fabsf(t3 - r[3]));
        }
    }
    float tp = block_sum_wmma(posf, s_part);
    float tr = block_sum_wmma(rsum, s_part);
    if (tid == 0) {
        atomicAdd(&npos[b], tp);
        atomicAdd(&reg_sum[b], tr);
    }
}

__device__ __forceinline__ float focal_elem(float x, int s, int c) {
    float p = fminf(fmaxf(x, 1e-4f), 1.0f - 1e-4f);
    if (s == c) {                       // positive anchor at its label
        float q = 1.0f - p;
        return 0.25f * q * q * (-__logf(p));
    }
    return 0.75f * p * p * (-__logf(1.0f - p));
}

// Vectorized focal loss: grid.y = image, grid-stride over A*C/4 float4s.
// Requires C % 4 == 0 (true for reference C=80). Index math is incremental
// (no per-iteration integer division) to stay under the HBM-implied VALU budget.
__global__ __launch_bounds__(TPB)
void cls_loss_kernel(const float* __restrict__ cls,
                     const int* __restrict__ state,
                     float* __restrict__ cls_sum,
                     int A, int C) {
    __shared__ float s_part[TPB];
    const int b = blockIdx.y;
    const int n4 = (A * C) >> 2;
    const float4* __restrict__ p4 = (const float4*)(cls + (size_t)b * A * C);
    const int* __restrict__ st = state + (size_t)b * A;

    const int stride4 = gridDim.x * TPB;
    const int i0 = blockIdx.x * TPB + threadIdx.x;
    int a  = (i0 * 4) / C;
    int c0 = (i0 * 4) - a * C;
    const int stride_e = stride4 * 4;
    const int da = stride_e / C;
    const int dc = stride_e - da * C;

    float acc = 0.0f;
    for (int i = i0; i < n4; i += stride4) {
        float4 v = p4[i];
        int s = st[a];
        if (s != -2) {  // ignored anchors contribute nothing
            acc += focal_elem(v.x, s, c0);
            acc += focal_elem(v.y, s, c0 + 1);
            acc += focal_elem(v.z, s, c0 + 2);
            acc += focal_elem(v.w, s, c0 + 3);
        }
        a += da; c0 += dc;
        if (c0 >= C) { c0 -= C; ++a; }
    }
    float tot = block_sum_wmma(acc, s_part);
    if (threadIdx.x == 0) atomicAdd(&cls_sum[b], tot);
}

// Scalar fallback for C % 4 != 0 (not hit with reference shapes).
__global__ __launch_bounds__(TPB)
void cls_loss_scalar_kernel(const float* __restrict__ cls,
                            const int* __restrict__ state,
                            float* __restrict__ cls_sum,
                            int A, int C) {
    __shared__ float s_part[TPB];
    const int b = blockIdx.y;
    const int n = A * C;
    const float* __restrict__ p = cls + (size_t)b * A * C;
    const int* __restrict__ st = state + (size_t)b * A;
    float acc = 0.0f;
    for (int i = blockIdx.x * TPB + threadIdx.x; i < n; i += gridDim.x * TPB) {
        int a = i / C, c = i - a * C;
        int s = st[a];
        if (s != -2) acc += focal_elem(p[i], s, c);
    }
    float tot = block_sum_wmma(acc, s_part);
    if (threadIdx.x == 0) atomicAdd(&cls_sum[b], tot);
}

__global__ void finalize_kernel(const float* __restrict__ cls_sum,
                                const float* __restrict__ reg_sum,
                                const float* __restrict__ npos,
                                float* __restrict__ out, int B) {
    if (blockIdx.x == 0 && threadIdx.x == 0) {
        float cm = 0.0f, rm = 0.0f;
        for (int b = 0; b < B; ++b) {
            float np = npos[b];
            cm += cls_sum[b] / fmaxf(np, 1.0f);
            rm += (np > 0.0f) ? reg_sum[b] / fmaxf(4.0f * np, 1.0f) : 0.0f;
        }
        out[0] = cm / (float)B;
        out[1] = rm / (float)B;
    }
}

extern "C" void kernel_launch(void* const* d_in, const int* in_sizes, int n_in,
                              void* d_out, int out_size, void* d_ws, size_t ws_size,
                              hipStream_t stream) {
    const float* cls     = (const float*)d_in[0];  // [B,A,C]
    const float* regs    = (const float*)d_in[1];  // [B,A,4]
    const float* anchors = (const float*)d_in[2];  // [1,A,4]
    const float* annots  = (const float*)d_in[3];  // [B,M,5]

    const int A = in_sizes[2] / 4;
    const int B = in_sizes[1] / (4 * A);
    const int C = in_sizes[0] / (B * A);
    int M = in_sizes[3] / (5 * B);
    if (M > MAXM) M = MAXM;

    // Workspace layout: [cls_sum[B] | reg_sum[B] | npos[B]] @0, state[B*A] @1024B.
    float* accf    = (float*)d_ws;
    float* cls_sum = accf;
    float* reg_sum = accf + B;
    float* npos    = accf + 2 * B;
    int*   state   = (int*)((char*)d_ws + 1024);

    zero_kernel<<<1, TPB, 0, stream>>>(accf, 3 * B);

    dim3 ga((A + TPB - 1) / TPB, B);
    assign_kernel<<<ga, TPB, 0, stream>>>(anchors, regs, annots, state,
                                          reg_sum, npos, A, M);

    if ((C & 3) == 0) {
        int n4 = (A * C) >> 2;
        int gx = (n4 + TPB * 8 - 1) / (TPB * 8);
        if (gx < 1) gx = 1;
        dim3 gc(gx, B);
        cls_loss_kernel<<<gc, TPB, 0, stream>>>(cls, state, cls_sum, A, C);
    } else {
        int n = A * C;
        int gx = (n + TPB * 16 - 1) / (TPB * 16);
        if (gx < 1) gx = 1;
        dim3 gc(gx, B);
        cls_loss_scalar_kernel<<<gc, TPB, 0, stream>>>(cls, state, cls_sum, A, C);
    }

    finalize_kernel<<<1, 32, 0, stream>>>(cls_sum, reg_sum, npos, (float*)d_out, B);
}